// PICModel5_32667521253815
// MI455X (gfx1250) — compile-verified
//
#include <hip/hip_runtime.h>
#include <cstddef>
#include <cstdint>

// ---------------------------------------------------------------------------
// Problem constants (from reference): H=768, D6=128, S=128, B=64, NL=3, V=81,
// LH=384, G=1536, BIA=100. Token count N_TOK = B*S = 8192.
// All math fp32 via V_WMMA_F32_16X16X4_F32 (keeps reference precision).
// ---------------------------------------------------------------------------
#define HDIM   768
#define D6DIM  128
#define SEQ    128
#define BATCH  64
#define NTOK   (BATCH * SEQ)
#define LH     384
#define GATES  1536
#define BIAPAD 112   // BIA=100 padded to multiple of 16 for tiling (ld only)

typedef __attribute__((ext_vector_type(2))) float v2f;
typedef __attribute__((ext_vector_type(8))) float v8f;

__device__ __forceinline__ v8f wmma4(v2f a, v2f b, v8f c) {
  // D(16x16 f32) = A(16x4 f32) x B(4x16 f32) + C ; wave32, EXEC all ones.
  return __builtin_amdgcn_wmma_f32_16x16x4_f32(false, a, false, b, (short)0, c,
                                               false, false);
}

__device__ __forceinline__ v8f v8zero() {
  v8f z;
#pragma unroll
  for (int i = 0; i < 8; ++i) z[i] = 0.0f;
  return z;
}

__device__ __forceinline__ float sigf(float x) {
  return 1.0f / (1.0f + __expf(-x));
}

// ---------------------------------------------------------------------------
// Generic fp32 WMMA GEMM:  C[M,N] (+)= A[M,K] * W[N,K]^T (+ bias + bias2)
// One wave owns a 16-row strip and all NT 16-col tiles (N <= NT*16).
// Software-pipelined: operands for step k+1 are prefetched into a second
// register set before the WMMAs of step k issue, so the per-iteration
// s_wait_loadcnt covers 1 wait per NT WMMAs and load latency hides under
// the matrix pipe.
// If span != nullptr, A rows are built on the fly:
//   row t (t = b*S + s): Ah[b*S + span[t,0]] + Ah[b*S + span[t,1]] + dep[span[t,2]]
// ---------------------------------------------------------------------------
template <int NT>
__global__ void gemm_wmma(const float* __restrict__ A, const float* __restrict__ W,
                          const float* __restrict__ bias,
                          const float* __restrict__ bias2, float* __restrict__ C,
                          int M, int N, int K, int lda, int ldw, int ldc,
                          int accumulate, const int* __restrict__ span,
                          const float* __restrict__ dep, int Sdim) {
  const int wave = threadIdx.x >> 5;
  const int lane = threadIdx.x & 31;
  const int lrow = lane & 15;
  const int lhi  = lane >> 4;
  const int mi = blockIdx.x * (blockDim.x >> 5) + wave;
  const int m0 = mi * 16;
  if (m0 >= M) return;  // wave-uniform

  const int mrow = m0 + lrow;
  const float *ar0, *ar1, *ar2;
  const bool use_span = (span != nullptr);
  if (use_span) {
    const int b  = mrow / Sdim;
    const int i0 = span[mrow * 3 + 0];
    const int i1 = span[mrow * 3 + 1];
    const int i2 = span[mrow * 3 + 2];
    ar0 = A + (size_t)(b * Sdim + i0) * lda;
    ar1 = A + (size_t)(b * Sdim + i1) * lda;
    ar2 = dep + (size_t)i2 * lda;
  } else {
    ar0 = A + (size_t)mrow * lda;
    ar1 = nullptr;
    ar2 = nullptr;
  }

  const float* wr[NT];
#pragma unroll
  for (int t = 0; t < NT; ++t) {
    int n = t * 16 + lrow;
    if (n >= N) n = N - 1;  // clamp; guarded at store
    wr[t] = W + (size_t)n * ldw;
  }

  v8f acc[NT];
#pragma unroll
  for (int t = 0; t < NT; ++t) acc[t] = v8zero();

  // ---- software pipeline: preload k=0 ----
  v2f a0, b0[NT];
  {
    const int ka = 2 * lhi;
    if (use_span) {
      a0[0] = ar0[ka] + ar1[ka] + ar2[ka];
      a0[1] = ar0[ka + 1] + ar1[ka + 1] + ar2[ka + 1];
    } else {
      a0 = *(const v2f*)(ar0 + ka);
    }
#pragma unroll
    for (int t = 0; t < NT; ++t) b0[t] = *(const v2f*)(wr[t] + ka);
  }

  for (int k = 0; k < K; k += 4) {
    const int kn = k + 4;
    v2f a1, b1[NT];
    if (kn < K) {  // wave-uniform
      const int ka = kn + 2 * lhi;
      if (use_span) {
        a1[0] = ar0[ka] + ar1[ka] + ar2[ka];
        a1[1] = ar0[ka + 1] + ar1[ka + 1] + ar2[ka + 1];
      } else {
        a1 = *(const v2f*)(ar0 + ka);
      }
#pragma unroll
      for (int t = 0; t < NT; ++t) b1[t] = *(const v2f*)(wr[t] + ka);
    }
#pragma unroll
    for (int t = 0; t < NT; ++t) acc[t] = wmma4(a0, b0[t], acc[t]);
    a0 = a1;
#pragma unroll
    for (int t = 0; t < NT; ++t) b0[t] = b1[t];
  }

#pragma unroll
  for (int t = 0; t < NT; ++t) {
    const int n = t * 16 + lrow;
    if (n < N) {
      float bv = 0.0f;
      if (bias) bv += bias[n];
      if (bias2) bv += bias2[n];
#pragma unroll
      for (int v = 0; v < 8; ++v) {
        const int m = m0 + v + 8 * lhi;
        float val = acc[t][v] + bv;
        float* p = C + (size_t)m * ldc + n;
        if (accumulate) val += *p;
        *p = val;
      }
    }
  }
}

// ---------------------------------------------------------------------------
// Bilinear core: Cout[t, o] += sum_i Si[t,i] * (sum_j Wb[o,i,j] * Aj[t,j])
// One wave per (16-token tile, channel o). U = Aj * Wb[o]^T via pipelined
// WMMA (Wb[o] rows i are contiguous over j), then per-row dot with Si
// reduced with __shfl_xor across the 16 lanes of each wave half.
// ---------------------------------------------------------------------------
__global__ void bilinear_wmma(const float* __restrict__ Aj,  // (NTOK,128) "ph"
                              const float* __restrict__ Si,  // (NTOK,128) "ps"
                              const float* __restrict__ Wb,  // (100,128,128)
                              float* __restrict__ Cout, int ldo) {
  const int wave = threadIdx.x >> 5;
  const int lane = threadIdx.x & 31;
  const int lrow = lane & 15;
  const int lhi  = lane >> 4;
  const int mi = blockIdx.x * 4 + wave;  // 512 token tiles
  const int o  = blockIdx.y;             // 100 channels
  const int m0 = mi * 16;

  const float* wrow = Wb + (size_t)o * D6DIM * D6DIM + (size_t)lrow * D6DIM;
  const float* arow = Aj + (size_t)(m0 + lrow) * D6DIM;

  v8f acc[8];
#pragma unroll
  for (int t = 0; t < 8; ++t) acc[t] = v8zero();

  v2f a0, b0[8];
  {
    const int ka = 2 * lhi;
    a0 = *(const v2f*)(arow + ka);
#pragma unroll
    for (int t = 0; t < 8; ++t) b0[t] = *(const v2f*)(wrow + (size_t)t * 16 * D6DIM + ka);
  }
  for (int k = 0; k < D6DIM; k += 4) {
    const int kn = k + 4;
    v2f a1, b1[8];
    if (kn < D6DIM) {
      const int ka = kn + 2 * lhi;
      a1 = *(const v2f*)(arow + ka);
#pragma unroll
      for (int t = 0; t < 8; ++t) b1[t] = *(const v2f*)(wrow + (size_t)t * 16 * D6DIM + ka);
    }
#pragma unroll
    for (int t = 0; t < 8; ++t) acc[t] = wmma4(a0, b0[t], acc[t]);
    a0 = a1;
#pragma unroll
    for (int t = 0; t < 8; ++t) b0[t] = b1[t];
  }

  float part[8];
#pragma unroll
  for (int v = 0; v < 8; ++v) part[v] = 0.0f;
#pragma unroll
  for (int t = 0; t < 8; ++t) {
    const int icol = t * 16 + lrow;
#pragma unroll
    for (int v = 0; v < 8; ++v) {
      const int m = m0 + v + 8 * lhi;
      part[v] += acc[t][v] * Si[(size_t)m * D6DIM + icol];
    }
  }
#pragma unroll
  for (int v = 0; v < 8; ++v) {
    float x = part[v];
#pragma unroll
    for (int off = 1; off < 16; off <<= 1) x += __shfl_xor(x, off, 32);
    if (lrow == 0) {
      const int m = m0 + v + 8 * lhi;
      Cout[(size_t)m * ldo + o] += x;  // single-owner RMW on top of linear part
    }
  }
}

// ---------------------------------------------------------------------------
// Sequential BiLSTM chains. One block (32 waves) per chain; hidden state h
// double-buffered in LDS (2 x 64 x 384 f32 = 192 KB of the 320 KB WGP LDS).
// Per step each wave runs 3 of the 96 (m-tile, u-tile) tasks; each task
// accumulates 4 gate tiles (i,f,g,o) over K = 100 (x @ w_ih^T, fused) +
// K = 384 (h @ w_hh^T) with pipelined f32 WMMA, then updates the cell
// state kept in VGPRs across all 128 steps.
// ---------------------------------------------------------------------------
struct ChainArgs {
  const float* bi;   // (B,S,BIAPAD), token index b*S+s
  const float* wih;  // (1536,100)
  const float* whh;  // (1536,384)
  const float* bih;  // (1536)
  const float* bhh;  // (1536)
  float* out;        // (64,768) fin buffer
  int outOff;        // 0 fwd, 384 bwd
  int rev;           // 0 fwd, 1 bwd
};
struct LstmPack { ChainArgs c[4]; };

__global__ __launch_bounds__(1024) void lstm_chain(LstmPack P, int S, int ldbi) {
  extern __shared__ float hbuf[];  // 2 * 64*384
  const ChainArgs ca = P.c[blockIdx.x];
  const int tid  = threadIdx.x;
  const int wave = tid >> 5;
  const int lane = tid & 31;
  const int lrow = lane & 15;
  const int lhi  = lane >> 4;
  const int HB = BATCH * LH;  // 24576

  for (int i = tid; i < HB; i += blockDim.x) hbuf[i] = 0.0f;  // h0 = 0
  __syncthreads();

  int mi[3], ui[3];
  float bI[3], bF[3], bG[3], bO[3];
#pragma unroll
  for (int tk = 0; tk < 3; ++tk) {
    const int task = wave * 3 + tk;  // 0..95
    mi[tk] = task / 24;              // 4 batch tiles
    ui[tk] = task % 24;              // 24 hidden tiles
    const int col = ui[tk] * 16 + lrow;
    bI[tk] = ca.bih[0 * LH + col] + ca.bhh[0 * LH + col];
    bF[tk] = ca.bih[1 * LH + col] + ca.bhh[1 * LH + col];
    bG[tk] = ca.bih[2 * LH + col] + ca.bhh[2 * LH + col];
    bO[tk] = ca.bih[3 * LH + col] + ca.bhh[3 * LH + col];
  }
  v8f creg[3];
#pragma unroll
  for (int tk = 0; tk < 3; ++tk) creg[tk] = v8zero();

  int p = 0;
  for (int st = 0; st < S; ++st) {
    const int s = ca.rev ? (S - 1 - st) : st;
    const float* hin = hbuf + p * HB;
    float* hout = hbuf + (1 - p) * HB;
#pragma unroll
    for (int tk = 0; tk < 3; ++tk) {
      const int m0 = mi[tk] * 16;
      const int u0 = ui[tk] * 16;
      v8f aI = v8zero(), aF = v8zero(), aG = v8zero(), aO = v8zero();

      // ---- x part: K = 100 (w_ih rows contiguous, ld = 100) ----
      {
        const float* arow = ca.bi + ((size_t)(m0 + lrow) * S + s) * ldbi;
        const float* wI = ca.wih + (size_t)(0 * LH + u0 + lrow) * 100;
        const float* wF = ca.wih + (size_t)(1 * LH + u0 + lrow) * 100;
        const float* wG = ca.wih + (size_t)(2 * LH + u0 + lrow) * 100;
        const float* wO = ca.wih + (size_t)(3 * LH + u0 + lrow) * 100;
        v2f a0, i0, f0, g0, o0;
        {
          const int ka = 2 * lhi;
          a0 = *(const v2f*)(arow + ka);
          i0 = *(const v2f*)(wI + ka);
          f0 = *(const v2f*)(wF + ka);
          g0 = *(const v2f*)(wG + ka);
          o0 = *(const v2f*)(wO + ka);
        }
        for (int k = 0; k < 100; k += 4) {
          const int kn = k + 4;
          v2f a1, i1, f1, g1, o1;
          if (kn < 100) {
            const int ka = kn + 2 * lhi;
            a1 = *(const v2f*)(arow + ka);
            i1 = *(const v2f*)(wI + ka);
            f1 = *(const v2f*)(wF + ka);
            g1 = *(const v2f*)(wG + ka);
            o1 = *(const v2f*)(wO + ka);
          }
          aI = wmma4(a0, i0, aI);
          aF = wmma4(a0, f0, aF);
          aG = wmma4(a0, g0, aG);
          aO = wmma4(a0, o0, aO);
          a0 = a1; i0 = i1; f0 = f1; g0 = g1; o0 = o1;
        }
      }
      // ---- h part: K = 384 (h read from LDS, w_hh from L2) ----
      {
        const float* hrow = hin + (size_t)(m0 + lrow) * LH;
        const float* vI = ca.whh + (size_t)(0 * LH + u0 + lrow) * LH;
        const float* vF = ca.whh + (size_t)(1 * LH + u0 + lrow) * LH;
        const float* vG = ca.whh + (size_t)(2 * LH + u0 + lrow) * LH;
        const float* vO = ca.whh + (size_t)(3 * LH + u0 + lrow) * LH;
        v2f a0, i0, f0, g0, o0;
        {
          const int ka = 2 * lhi;
          a0 = *(const v2f*)(hrow + ka);
          i0 = *(const v2f*)(vI + ka);
          f0 = *(const v2f*)(vF + ka);
          g0 = *(const v2f*)(vG + ka);
          o0 = *(const v2f*)(vO + ka);
        }
        for (int k = 0; k < LH; k += 4) {
          const int kn = k + 4;
          v2f a1, i1, f1, g1, o1;
          if (kn < LH) {
            const int ka = kn + 2 * lhi;
            a1 = *(const v2f*)(hrow + ka);
            i1 = *(const v2f*)(vI + ka);
            f1 = *(const v2f*)(vF + ka);
            g1 = *(const v2f*)(vG + ka);
            o1 = *(const v2f*)(vO + ka);
          }
          aI = wmma4(a0, i0, aI);
          aF = wmma4(a0, f0, aF);
          aG = wmma4(a0, g0, aG);
          aO = wmma4(a0, o0, aO);
          a0 = a1; i0 = i1; f0 = f1; g0 = g1; o0 = o1;
        }
      }
      // gate nonlinearity + cell update; C layout: [m0+v+8*lhi, u0+lrow]
      const int ucol = u0 + lrow;
#pragma unroll
      for (int v = 0; v < 8; ++v) {
        const float gi = sigf(aI[v] + bI[tk]);
        const float gf = sigf(aF[v] + bF[tk]);
        const float gg = tanhf(aG[v] + bG[tk]);
        const float go = sigf(aO[v] + bO[tk]);
        const float c = gf * creg[tk][v] + gi * gg;
        creg[tk][v] = c;
        const int m = m0 + v + 8 * lhi;
        hout[(size_t)m * LH + ucol] = go * tanhf(c);
      }
    }
    __syncthreads();
    p ^= 1;
  }

  const float* hf = hbuf + p * HB;
  for (int i = tid; i < HB; i += blockDim.x) {
    const int b = i / LH, u = i % LH;
    ca.out[(size_t)b * (2 * LH) + ca.outOff + u] = hf[i];
  }
}

// ---------------------------------------------------------------------------
// Final tiny bilinear: out[b,l] = pf[b]·Wb[l]·hf[b] + pf·l1w[l] + l1b[l]
//                                 + hf·l2w[l] + l2b[l].   (64 x 3 output)
// ---------------------------------------------------------------------------
__global__ void final_bilinear(const float* __restrict__ pf,
                               const float* __restrict__ hf,
                               const float* __restrict__ Wb,   // (3,768,768)
                               const float* __restrict__ l1w,  // (3,768)
                               const float* __restrict__ l1b,
                               const float* __restrict__ l2w,
                               const float* __restrict__ l2b,
                               float* __restrict__ out) {
  __shared__ float hsm[HDIM];
  __shared__ float red[256];
  const int b = blockIdx.x, l = blockIdx.y, tid = threadIdx.x;
  for (int j = tid; j < HDIM; j += 256) hsm[j] = hf[(size_t)b * HDIM + j];
  __syncthreads();
  const float* W = Wb + (size_t)l * HDIM * HDIM;
  float part = 0.0f;
  for (int i = tid; i < HDIM; i += 256) {
    const float* wr = W + (size_t)i * HDIM;
    float dot = 0.0f;
    for (int j = 0; j < HDIM; ++j) dot += wr[j] * hsm[j];
    part += pf[(size_t)b * HDIM + i] * (dot + l1w[(size_t)l * HDIM + i]);
    part += hsm[i] * l2w[(size_t)l * HDIM + i];
  }
  red[tid] = part;
  __syncthreads();
  for (int sdx = 128; sdx > 0; sdx >>= 1) {
    if (tid < sdx) red[tid] += red[tid + sdx];
    __syncthreads();
  }
  if (tid == 0) out[b * 3 + l] = red[0] + l1b[l] + l2b[l];
}

// ---------------------------------------------------------------------------
// Host launcher. Input order = setup_inputs() dict order, params flattened in
// insertion order (nested lstm dicts recursively).
// ---------------------------------------------------------------------------
extern "C" void kernel_launch(void* const* d_in, const int* in_sizes, int n_in,
                              void* d_out, int out_size, void* d_ws,
                              size_t ws_size, hipStream_t stream) {
  const float* hidden    = (const float*)d_in[0];   // (2,B,S,H)
  const int*   prem_span = (const int*)d_in[1];     // (B,S,3)
  const int*   hypo_span = (const int*)d_in[2];     // (B,S,3)
  // d_in[3] = batch_size scalar (unused; B fixed at 64)
  const float* dep_emb = (const float*)d_in[4];     // (81,768)
  const float* red1_w = (const float*)d_in[5];  const float* red1_b = (const float*)d_in[6];
  const float* red2_w = (const float*)d_in[7];  const float* red2_b = (const float*)d_in[8];
  const float* red3_w = (const float*)d_in[9];  const float* red3_b = (const float*)d_in[10];
  const float* red4_w = (const float*)d_in[11]; const float* red4_b = (const float*)d_in[12];
  const float* w1_bil  = (const float*)d_in[13];
  const float* w1_l1_w = (const float*)d_in[14]; const float* w1_l1_b = (const float*)d_in[15];
  const float* w1_l2_w = (const float*)d_in[16]; const float* w1_l2_b = (const float*)d_in[17];
  const float* w2_bil  = (const float*)d_in[18];
  const float* w2_l1_w = (const float*)d_in[19]; const float* w2_l1_b = (const float*)d_in[20];
  const float* w2_l2_w = (const float*)d_in[21]; const float* w2_l2_b = (const float*)d_in[22];
  const float* l1_wih_f = (const float*)d_in[23]; const float* l1_whh_f = (const float*)d_in[24];
  const float* l1_bih_f = (const float*)d_in[25]; const float* l1_bhh_f = (const float*)d_in[26];
  const float* l1_wih_b = (const float*)d_in[27]; const float* l1_whh_b = (const float*)d_in[28];
  const float* l1_bih_b = (const float*)d_in[29]; const float* l1_bhh_b = (const float*)d_in[30];
  const float* l2_wih_f = (const float*)d_in[31]; const float* l2_whh_f = (const float*)d_in[32];
  const float* l2_bih_f = (const float*)d_in[33]; const float* l2_bhh_f = (const float*)d_in[34];
  const float* l2_wih_b = (const float*)d_in[35]; const float* l2_whh_b = (const float*)d_in[36];
  const float* l2_bih_b = (const float*)d_in[37]; const float* l2_bhh_b = (const float*)d_in[38];
  const float* bia_bil  = (const float*)d_in[39];
  const float* bia_l1_w = (const float*)d_in[40]; const float* bia_l1_b = (const float*)d_in[41];
  const float* bia_l2_w = (const float*)d_in[42]; const float* bia_l2_b = (const float*)d_in[43];
  (void)in_sizes; (void)n_in; (void)out_size; (void)ws_size;

  const float* prem_h = hidden;
  const float* hypo_h = hidden + (size_t)BATCH * SEQ * HDIM;

  // Workspace layout (fp32), total ~24.5 MB.
  float* ws  = (float*)d_ws;
  float* ps  = ws;                          // NTOK*128
  float* ph  = ps  + (size_t)NTOK * D6DIM;
  float* hsb = ph  + (size_t)NTOK * D6DIM;
  float* hhb = hsb + (size_t)NTOK * D6DIM;
  float* pbi = hhb + (size_t)NTOK * D6DIM;  // NTOK*BIAPAD
  float* hbi = pbi + (size_t)NTOK * BIAPAD;
  float* pfin = hbi + (size_t)NTOK * BIAPAD;  // 64*768
  float* hfin = pfin + (size_t)BATCH * 2 * LH;

  const dim3 gblk(128);              // 4 waves/block
  const int  ggrd = NTOK / 16 / 4;   // 128 blocks for M=8192

  // Stage B: reductions (span gather fused into A-loads for ps/hs).
  gemm_wmma<8><<<ggrd, gblk, 0, stream>>>(prem_h, red1_w, red1_b, nullptr, ps,
      NTOK, D6DIM, HDIM, HDIM, HDIM, D6DIM, 0, prem_span, dep_emb, SEQ);
  gemm_wmma<8><<<ggrd, gblk, 0, stream>>>(prem_h, red2_w, red2_b, nullptr, ph,
      NTOK, D6DIM, HDIM, HDIM, HDIM, D6DIM, 0, nullptr, nullptr, SEQ);
  gemm_wmma<8><<<ggrd, gblk, 0, stream>>>(hypo_h, red3_w, red3_b, nullptr, hsb,
      NTOK, D6DIM, HDIM, HDIM, HDIM, D6DIM, 0, hypo_span, dep_emb, SEQ);
  gemm_wmma<8><<<ggrd, gblk, 0, stream>>>(hypo_h, red4_w, red4_b, nullptr, hhb,
      NTOK, D6DIM, HDIM, HDIM, HDIM, D6DIM, 0, nullptr, nullptr, SEQ);

  // Stage C: linear terms into bi buffers (N=100, NT=7), then bilinear adds.
  gemm_wmma<7><<<ggrd, gblk, 0, stream>>>(ps, w1_l1_w, w1_l1_b, w1_l2_b, pbi,
      NTOK, 100, D6DIM, D6DIM, D6DIM, BIAPAD, 0, nullptr, nullptr, SEQ);
  gemm_wmma<7><<<ggrd, gblk, 0, stream>>>(ph, w1_l2_w, nullptr, nullptr, pbi,
      NTOK, 100, D6DIM, D6DIM, D6DIM, BIAPAD, 1, nullptr, nullptr, SEQ);
  gemm_wmma<7><<<ggrd, gblk, 0, stream>>>(hsb, w2_l1_w, w2_l1_b, w2_l2_b, hbi,
      NTOK, 100, D6DIM, D6DIM, D6DIM, BIAPAD, 0, nullptr, nullptr, SEQ);
  gemm_wmma<7><<<ggrd, gblk, 0, stream>>>(hhb, w2_l2_w, nullptr, nullptr, hbi,
      NTOK, 100, D6DIM, D6DIM, D6DIM, BIAPAD, 1, nullptr, nullptr, SEQ);

  bilinear_wmma<<<dim3(128, 100), gblk, 0, stream>>>(ph, ps, w1_bil, pbi, BIAPAD);
  bilinear_wmma<<<dim3(128, 100), gblk, 0, stream>>>(hhb, hsb, w2_bil, hbi, BIAPAD);

  // Stage D: 4 sequential BiLSTM chains, one WGP-resident block each.
  LstmPack P;
  P.c[0] = {pbi, l1_wih_f, l1_whh_f, l1_bih_f, l1_bhh_f, pfin, 0,  0};
  P.c[1] = {pbi, l1_wih_b, l1_whh_b, l1_bih_b, l1_bhh_b, pfin, LH, 1};
  P.c[2] = {hbi, l2_wih_f, l2_whh_f, l2_bih_f, l2_bhh_f, hfin, 0,  0};
  P.c[3] = {hbi, l2_wih_b, l2_whh_b, l2_bih_b, l2_bhh_b, hfin, LH, 1};
  lstm_chain<<<4, 1024, 2 * BATCH * LH * sizeof(float), stream>>>(P, SEQ, BIAPAD);

  // Stage E: final 64x3 bilinear head.
  final_bilinear<<<dim3(BATCH, 3), 256, 0, stream>>>(
      pfin, hfin, bia_bil, bia_l1_w, bia_l1_b, bia_l2_w, bia_l2_b, (float*)d_out);
}